// MaskRCNN_53395033424377
// MI455X (gfx1250) — compile-verified
//
#include <hip/hip_runtime.h>
#include <cstdint>
#include <cstddef>

#define N_ROIS        6000
#define NUM_CLASSES   81
#define MIN_CONF      0.7f
#define NMS_THRESH    0.3f
#define MAX_INST      100
#define SORT_N        8192
#define MASK_WORDS    192      // 6144 padded columns / 32
#define CHUNK         512

typedef unsigned int u32x4 __attribute__((ext_vector_type(4)));
typedef unsigned int u32x8 __attribute__((ext_vector_type(8)));

// ---------------------------------------------------------------------------
// Kernel 1: per-ROI argmax over 81 classes (one wave32 per ROI), apply class-
// specific deltas, scale to 1024x1024, clip to window, round-half-even.
// ---------------------------------------------------------------------------
__global__ __launch_bounds__(256)
void refine_kernel(const float* __restrict__ rois,
                   const float* __restrict__ probs,
                   const float* __restrict__ deltas,
                   const float* __restrict__ window,
                   float4* __restrict__ refined,
                   float* __restrict__ scores,
                   int* __restrict__ cls) {
  const int warp = threadIdx.x >> 5;
  const int lane = threadIdx.x & 31;
  const int roi  = blockIdx.x * 8 + warp;
  if (roi >= N_ROIS) return;

  // 81 classes on 32 lanes: two full rounds + predicated tail; within a lane
  // strict '>' keeps the first (lowest class) on ties.
  const int base = roi * NUM_CLASSES;
  float best = probs[base + lane];
  int   bidx = lane;
  {
    const float v1 = probs[base + 32 + lane];
    if (v1 > best) { best = v1; bidx = 32 + lane; }
    if (64 + lane < NUM_CLASSES) {
      const float v2 = probs[base + 64 + lane];
      if (v2 > best) { best = v2; bidx = 64 + lane; }
    }
  }
  // butterfly reduction, tie-break to lower class index (jnp.argmax semantics)
  for (int off = 16; off > 0; off >>= 1) {
    const float ov = __shfl_xor(best, off, 32);
    const int   oi = __shfl_xor(bidx, off, 32);
    if (ov > best || (ov == best && oi < bidx)) { best = ov; bidx = oi; }
  }

  if (lane == 0) {
    const float4 box = ((const float4*)rois)[roi];                   // y1,x1,y2,x2
    const float4 d   = ((const float4*)deltas)[roi * NUM_CLASSES + bidx];
    const float dy = d.x * 0.1f, dx = d.y * 0.1f;
    const float dh = d.z * 0.2f, dw = d.w * 0.2f;
    float h = box.z - box.x;
    float w = box.w - box.y;
    const float cy = box.x + 0.5f * h + dy * h;
    const float cx = box.y + 0.5f * w + dx * w;
    h = h * expf(dh);
    w = w * expf(dw);
    float y1 = cy - 0.5f * h, x1 = cx - 0.5f * w;
    float y2 = y1 + h,        x2 = x1 + w;
    y1 *= 1024.0f; x1 *= 1024.0f; y2 *= 1024.0f; x2 *= 1024.0f;
    const float wy1 = window[0], wx1 = window[1];
    const float wy2 = window[2], wx2 = window[3];
    y1 = rintf(fminf(fmaxf(y1, wy1), wy2));   // rintf == round half to even
    x1 = rintf(fminf(fmaxf(x1, wx1), wx2));
    y2 = rintf(fminf(fmaxf(y2, wy1), wy2));
    x2 = rintf(fminf(fmaxf(x2, wx1), wx2));
    refined[roi] = make_float4(y1, x1, y2, x2);
    const bool keep = (bidx > 0) && (best >= MIN_CONF);
    scores[roi] = keep ? best : -1.0f;
    cls[roi]    = bidx;
  }
}

// ---------------------------------------------------------------------------
// Kernel 2: single-workgroup bitonic sort of 8192 packed 64-bit keys in LDS.
// key = (~floatflip(score) << 32) | idx  -> ascending u64 sort == stable
// descending score order (ties broken by ascending original index), exactly
// matching jnp.argsort(-scores_masked). Pads with score=-2 (< -1 sentinel).
// ---------------------------------------------------------------------------
__global__ __launch_bounds__(1024)
void sort_kernel(const float* __restrict__ scores,
                 const float4* __restrict__ boxes,
                 const int* __restrict__ cls,
                 float4* __restrict__ sboxes,
                 float* __restrict__ sscores,
                 int* __restrict__ scls) {
  __shared__ unsigned long long key[SORT_N];   // 64 KB LDS
  const int tid = threadIdx.x;

  for (int i = tid; i < SORT_N; i += 1024) {
    const float sc = (i < N_ROIS) ? scores[i] : -2.0f;
    const unsigned b    = __float_as_uint(sc);
    const unsigned asc  = b ^ (unsigned)(((int)b >> 31) | 0x80000000);  // monotone ascending map
    const unsigned desc = ~asc;                                        // monotone descending map
    key[i] = ((unsigned long long)desc << 32) | (unsigned)i;
  }
  __syncthreads();

  for (int k = 2; k <= SORT_N; k <<= 1) {
    for (int j = k >> 1; j > 0; j >>= 1) {
      for (int e = 0; e < SORT_N / 1024; ++e) {
        const int i   = tid + e * 1024;
        const int ixj = i ^ j;
        if (ixj > i) {
          const unsigned long long a = key[i];
          const unsigned long long b = key[ixj];
          const bool up = ((i & k) == 0);
          if ((a > b) == up) { key[i] = b; key[ixj] = a; }
        }
      }
      __syncthreads();
    }
  }

  for (int i = tid; i < N_ROIS; i += 1024) {
    const int oi = (int)(key[i] & 0xFFFFFFFFull);
    sboxes[i]  = boxes[oi];
    sscores[i] = scores[oi];
    scls[i]    = cls[oi];
  }
}

// ---------------------------------------------------------------------------
// Kernel 3: suppression bitmask. One wave per sorted row; 512-box column
// chunks (8 KB) staged into LDS by the Tensor Data Mover: wave 0 builds a
// 2-group D# (count=1, data_size=8B, 1-line tile of 1024 units, tensor_dim0 =
// remaining valid units so OOB reads zero-fill the tail) and issues
// tensor_load_to_lds, then waits TENSORcnt before the barrier publishes the
// tile to the block's 8 waves. Class equality is exactly equivalent to the
// reference's per-class coordinate offset (translation-invariant IoU);
// 0/0 -> NaN -> (NaN > 0.3) == false matches JAX. wave32 __ballot packs 32
// column predicates into one mask word.
// ---------------------------------------------------------------------------
__global__ __launch_bounds__(256)
void iou_mask_kernel(const float4* __restrict__ sboxes,
                     const int* __restrict__ scls,
                     unsigned* __restrict__ mask) {
  __shared__ float4 cb[CHUNK];
  __shared__ int    cc[CHUNK];
  const int warp = threadIdx.x >> 5;
  const int lane = threadIdx.x & 31;
  const int row  = blockIdx.x * 8 + warp;
  const bool rowok = (row < N_ROIS);

  float4 bi = make_float4(0.f, 0.f, 0.f, 0.f);
  float areai = 0.0f;
  int   ci = -2;
  if (rowok) {
    bi = sboxes[row];
    ci = scls[row];
    areai = (bi.z - bi.x) * (bi.w - bi.y);
  }

  for (int c0 = 0; c0 < MASK_WORDS * 32; c0 += CHUNK) {
    __syncthreads();               // previous chunk fully consumed
    if (warp == 0) {
      // --- TDM: DMA 512 float4 boxes (1024 x 8B units) global -> LDS ---
      const unsigned long long ga = (unsigned long long)(uintptr_t)(sboxes + c0);
      const unsigned ldsa = (unsigned)(uintptr_t)(&cb[0]); // low 32b = DS offset
      const unsigned td0  = (unsigned)(N_ROIS - c0) * 2u;  // valid 8B units left
      const u32x4 g0 = {
        1u,                                                  // count=1 (valid D#)
        ldsa,                                                // lds_addr
        (unsigned)ga,                                        // global_addr[31:0]
        (((unsigned)(ga >> 32)) & 0x01FFFFFFu) | 0x80000000u // addr[56:32]|type=2
      };
      const u32x8 g1 = {
        0x00030000u,                         // wg_mask=0, data_size=3 (8B)
        (td0 & 0xFFFFu) << 16,               // tensor_dim0[15:0]
        ((td0 >> 16) & 0xFFFFu) | (1u << 16),// tensor_dim0[31:16], tensor_dim1=1
        (unsigned)(CHUNK * 2) << 16,         // tile_dim0 = 1024 units
        1u,                                  // tile_dim1 = 1, tile_dim2 = 0
        td0,                                 // tensor_dim0_stride[31:0]
        0u,                                  // stride[47:32]=0, dim1_stride lo
        0u
      };
      asm volatile("tensor_load_to_lds %0, %1" :: "s"(g0), "s"(g1) : "memory");
      __builtin_amdgcn_s_wait_tensorcnt(0);
    }
    // classes (+ -1 padding) via regular path; prefetch next chunk's classes
    for (int t = threadIdx.x; t < CHUNK; t += 256) {
      const int j = c0 + t;
      cc[t] = (j < N_ROIS) ? scls[j] : -1;
    }
    if (c0 + CHUNK < N_ROIS)
      __builtin_prefetch(scls + c0 + CHUNK + threadIdx.x, 0, 1);
    __syncthreads();
    if (rowok) {
      for (int ww = 0; ww < CHUNK / 32; ++ww) {
        const int    j  = c0 + ww * 32 + lane;
        const float4 bj = cb[ww * 32 + lane];
        const int    cj = cc[ww * 32 + lane];
        bool pred = false;
        if (j > row && cj == ci) {
          const float yy1 = fmaxf(bi.x, bj.x), xx1 = fmaxf(bi.y, bj.y);
          const float yy2 = fminf(bi.z, bj.z), xx2 = fminf(bi.w, bj.w);
          const float inter = fmaxf(xx2 - xx1, 0.0f) * fmaxf(yy2 - yy1, 0.0f);
          const float areaj = (bj.z - bj.x) * (bj.w - bj.y);
          const float iou = inter / (areai + areaj - inter);
          pred = iou > NMS_THRESH;
        }
        const unsigned bits = (unsigned)__ballot(pred);    // wave32: 32 valid bits
        if (lane == 0) mask[(size_t)row * MASK_WORDS + (c0 >> 5) + ww] = bits;
      }
    }
  }
}

// ---------------------------------------------------------------------------
// Kernel 4: greedy sequential NMS scan + emit. Single wave32. The "removed"
// bitmask (6144 bits) lives in registers: 6 words per lane. For each kept row
// its 192 mask words are DMA'd into LDS with the CDNA5 async-to-LDS path
// (GLOBAL_LOAD_ASYNC_TO_LDS_B32, tracked by ASYNCcnt) then OR'd in.
// Kept entries appear in descending-score order, so the first 100 kept are
// exactly top_k(final_scores, 100); remaining rows are zero (det * valid).
// ---------------------------------------------------------------------------
__global__ __launch_bounds__(32)
void nms_scan_emit(const unsigned* __restrict__ mask,
                   const float* __restrict__ sscores,
                   const float4* __restrict__ sboxes,
                   const int* __restrict__ scls,
                   float* __restrict__ out) {
  __shared__ unsigned rowbuf[MASK_WORDS];
  __shared__ int      sel[MAX_INST];
  const int lane = threadIdx.x;   // 0..31

  unsigned r0 = 0, r1 = 0, r2 = 0, r3 = 0, r4 = 0, r5 = 0;
  int count = 0;

  for (int i = 0; i < N_ROIS; ++i) {
    if (count >= MAX_INST) break;
    const int w = i >> 5;                 // mask word holding bit for row i
    unsigned myw;
    switch (w >> 5) {                     // uniform: which register slot owns it
      case 0:  myw = r0; break;
      case 1:  myw = r1; break;
      case 2:  myw = r2; break;
      case 3:  myw = r3; break;
      case 4:  myw = r4; break;
      default: myw = r5; break;
    }
    const unsigned word = __shfl(myw, w & 31, 32);           // broadcast owner lane
    const bool removed  = (word >> (i & 31)) & 1u;
    const float sc = sscores[i];                              // uniform scalar load
    if (!removed && sc > 0.0f) {
      if (lane == 0) sel[count] = i;
      ++count;
      // async-copy mask row i (192 dwords, 6 per lane) into LDS
      const unsigned long long rowaddr =
          (unsigned long long)(uintptr_t)(mask + (size_t)i * MASK_WORDS);
#pragma unroll
      for (int s = 0; s < 6; ++s) {
        // low 32 bits of a generic LDS pointer == DS byte offset
        const unsigned lds_off =
            (unsigned)(uintptr_t)(&rowbuf[s * 32 + lane]);
        const unsigned long long ga = rowaddr + (unsigned)((s * 32 + lane) * 4);
        asm volatile("global_load_async_to_lds_b32 %0, %1, off"
                     :: "v"(lds_off), "v"(ga) : "memory");
      }
      asm volatile("s_wait_asynccnt 0" ::: "memory");
      r0 |= rowbuf[lane];
      r1 |= rowbuf[lane + 32];
      r2 |= rowbuf[lane + 64];
      r3 |= rowbuf[lane + 96];
      r4 |= rowbuf[lane + 128];
      r5 |= rowbuf[lane + 160];
    }
  }
  __syncthreads();

  // emit det[100,6], zero-padded past the last kept detection
  for (int k = lane; k < MAX_INST; k += 32) {
    float o0 = 0.f, o1 = 0.f, o2 = 0.f, o3 = 0.f, o4 = 0.f, o5 = 0.f;
    if (k < count) {
      const int i = sel[k];
      const float4 b = sboxes[i];
      o0 = b.x; o1 = b.y; o2 = b.z; o3 = b.w;
      o4 = (float)scls[i];
      o5 = sscores[i];
    }
    float* o = out + k * 6;
    o[0] = o0; o[1] = o1; o[2] = o2; o[3] = o3; o[4] = o4; o[5] = o5;
  }
}

// ---------------------------------------------------------------------------
// Host launcher. Workspace layout (all offsets 16B aligned, total ~4.9 MB):
//   [0      , 96000 )  refined boxes  float4[6000]
//   [96000  , 120000)  masked scores  float [6000]
//   [120000 , 144000)  class ids      int   [6000]
//   [144000 , 240000)  sorted boxes   float4[6000]
//   [240000 , 264000)  sorted scores  float [6000]
//   [264000 , 288000)  sorted classes int   [6000]
//   [288000 , 4896000) suppression mask u32[6000][192]
// ---------------------------------------------------------------------------
extern "C" void kernel_launch(void* const* d_in, const int* in_sizes, int n_in,
                              void* d_out, int out_size, void* d_ws, size_t ws_size,
                              hipStream_t stream) {
  (void)in_sizes; (void)n_in; (void)out_size; (void)ws_size;
  const float* rois    = (const float*)d_in[0];
  const float* probs   = (const float*)d_in[1];
  const float* deltas  = (const float*)d_in[2];
  const float* window  = (const float*)d_in[3];

  char* ws = (char*)d_ws;
  float4*   refined = (float4*)(ws + 0);
  float*    scores  = (float*)(ws + 96000);
  int*      cls     = (int*)(ws + 120000);
  float4*   sboxes  = (float4*)(ws + 144000);
  float*    sscores = (float*)(ws + 240000);
  int*      scls    = (int*)(ws + 264000);
  unsigned* mask    = (unsigned*)(ws + 288000);

  refine_kernel<<<750, 256, 0, stream>>>(rois, probs, deltas, window,
                                         refined, scores, cls);
  sort_kernel<<<1, 1024, 0, stream>>>(scores, refined, cls,
                                      sboxes, sscores, scls);
  iou_mask_kernel<<<750, 256, 0, stream>>>(sboxes, scls, mask);
  nms_scan_emit<<<1, 32, 0, stream>>>(mask, sscores, sboxes, scls,
                                      (float*)d_out);
}